// LocalInteraction_13005160972683
// MI455X (gfx1250) — compile-verified
//
#include <hip/hip_runtime.h>
#include <math.h>

typedef __attribute__((ext_vector_type(2))) float v2f;
typedef __attribute__((ext_vector_type(8))) float v8f;

constexpr int NA = 5000;
constexpr int NE = 100000;
constexpr int F  = 128;
constexpr int NT = (NA + 15) / 16;  // 313 atom tiles
constexpr int ST = 132;             // padded LDS row stride (bank-conflict free float2)

__device__ __constant__ float c_binom[16] = {
    1.f, 15.f, 105.f, 455.f, 1365.f, 3003.f, 5005.f, 6435.f,
    6435.f, 5005.f, 3003.f, 1365.f, 455.f, 105.f, 15.f, 1.f};

__device__ inline v8f wmma4(v2f a, v2f b, v8f c) {
  // D = A(16x4 f32) * B(4x16 f32) + C(16x16 f32), full-precision matrix op
  return __builtin_amdgcn_wmma_f32_16x16x4_f32(false, a, false, b, (short)0, c,
                                               false, false);
}

__device__ inline float swishf(float x, float a, float b) {
  return a * x / (1.f + expf(-b * x));
}

// ---------------------------------------------------------------------------
// 3-layer residual MLP:  out = [resid? x:0] + W3*sw2( x + W2*sw1(W1*sw0(x)+b1)+b2 ) + b3
// One block = 16 atoms, 8 waves; wave w computes output features [16w,16w+16).
// ---------------------------------------------------------------------------
__global__ __launch_bounds__(256) void mlp3_kernel(
    const float* __restrict__ X,
    const float* __restrict__ W1, const float* __restrict__ B1,
    const float* __restrict__ W2, const float* __restrict__ B2,
    const float* __restrict__ W3, const float* __restrict__ B3,
    const float* __restrict__ alpha, const float* __restrict__ beta,
    int sw0, int sw1, int sw2,
    float* __restrict__ Out,
    int wStride, int bStride, int outStride, int lastResid) {
  __shared__ __align__(16) float xs[16 * ST];
  __shared__ __align__(16) float bufA[16 * ST];
  __shared__ __align__(16) float bufB[16 * ST];

  const int tile = blockIdx.x;
  const int c = blockIdx.y;
  const int n0 = tile * 16;
  const float* w1 = W1 + c * wStride;
  const float* w2 = W2 + c * wStride;
  const float* w3 = W3 + c * wStride;
  const float* b1 = B1 + c * bStride;
  const float* b2 = B2 + c * bStride;
  const float* b3 = B3 + c * bStride;
  float* out = Out + c * outStride;

  const float a0 = alpha[sw0], e0 = beta[sw0];
  const float a1 = alpha[sw1], e1 = beta[sw1];
  const float a2 = alpha[sw2], e2 = beta[sw2];

  const int t = threadIdx.x;
  for (int idx = t; idx < 16 * F; idx += 256) {
    int r = idx >> 7, col = idx & 127;
    int n = n0 + r;
    float v = (n < NA) ? X[n * F + col] : 0.f;
    xs[r * ST + col] = v;
    bufA[r * ST + col] = swishf(v, a0, e0);
  }
  __syncthreads();

  const int wave = t >> 5;
  const int lane = t & 31;
  const int l15 = lane & 15;
  const int hi = lane >> 4;
  const int fcol = wave * 16 + l15;  // this lane's output feature (B column)
  const int koff = 2 * hi;

  // ---- layer 1: bufA -> sw1 -> bufB
  {
    float bias = b1[fcol];
    v8f acc;
#pragma unroll
    for (int r = 0; r < 8; r++) acc[r] = bias;
#pragma unroll 8
    for (int k0 = 0; k0 < F; k0 += 4) {
      v2f a = *(const v2f*)&bufA[l15 * ST + k0 + koff];
      v2f b = *(const v2f*)&w1[fcol * F + k0 + koff];
      acc = wmma4(a, b, acc);
    }
#pragma unroll
    for (int r = 0; r < 8; r++)
      bufB[(r + 8 * hi) * ST + fcol] = swishf(acc[r], a1, e1);
  }
  __syncthreads();

  // ---- layer 2: bufB -> h = xs + (.) ; bufA = sw2(h)
  {
    float bias = b2[fcol];
    v8f acc;
#pragma unroll
    for (int r = 0; r < 8; r++) acc[r] = bias;
#pragma unroll 8
    for (int k0 = 0; k0 < F; k0 += 4) {
      v2f a = *(const v2f*)&bufB[l15 * ST + k0 + koff];
      v2f b = *(const v2f*)&w2[fcol * F + k0 + koff];
      acc = wmma4(a, b, acc);
    }
#pragma unroll
    for (int r = 0; r < 8; r++) {
      float h = acc[r] + xs[(r + 8 * hi) * ST + fcol];
      bufA[(r + 8 * hi) * ST + fcol] = swishf(h, a2, e2);
    }
  }
  __syncthreads();

  // ---- layer 3: bufA -> out (+xs if residual)
  {
    float bias = b3[fcol];
    v8f acc;
#pragma unroll
    for (int r = 0; r < 8; r++) acc[r] = bias;
#pragma unroll 8
    for (int k0 = 0; k0 < F; k0 += 4) {
      v2f a = *(const v2f*)&bufA[l15 * ST + k0 + koff];
      v2f b = *(const v2f*)&w3[fcol * F + k0 + koff];
      acc = wmma4(a, b, acc);
    }
#pragma unroll
    for (int r = 0; r < 8; r++) {
      int n = n0 + r + 8 * hi;
      if (n < NA) {
        float v = acc[r] + (lastResid ? xs[(r + 8 * hi) * ST + fcol] : 0.f);
        out[n * F + fcol] = v;
      }
    }
  }
}

// ---------------------------------------------------------------------------
// Edge kernel: rho (Bernstein*cutoff) x Y (spherical harmonics) is rank-1 in
// (k,m), so mm[f,m] = (G @ rho)[f] * Y[m].  128 threads per edge, one feature
// per thread, scatter-add into q buffers with f32 atomics.
// q_s: (NA,F) row-major.  q_p: (F, 3*NA) feature-major.  q_d: (F, 5*NA).
// ---------------------------------------------------------------------------
__global__ __launch_bounds__(256) void edge_kernel(
    const float* __restrict__ xyz, const int* __restrict__ nbrs,
    const float* __restrict__ res,
    const float* __restrict__ Gs, const float* __restrict__ Gp,
    const float* __restrict__ Gd,
    const float* __restrict__ pgs, const float* __restrict__ pgp,
    const float* __restrict__ pgd,
    float* __restrict__ qs, float* __restrict__ qp, float* __restrict__ qd) {
  __shared__ float rho[2][3][16];
  __shared__ float ysh[2][8];
  const int half = threadIdx.x >> 7;
  const int f = threadIdx.x & 127;
  const int e = blockIdx.x * 2 + half;
  int i0 = 0, i1 = 0;
  if (e < NE) {
    i0 = nbrs[2 * e];
    i1 = nbrs[2 * e + 1];
  }
  if (e < NE && f < 56) {
    float dx = xyz[3 * i1 + 0] - xyz[3 * i0 + 0];
    float dy = xyz[3 * i1 + 1] - xyz[3 * i0 + 1];
    float dz = xyz[3 * i1 + 2] - xyz[3 * i0 + 2];
    float r2 = dx * dx + dy * dy + dz * dz + 3.0e-15f;
    float r = sqrtf(r2);
    if (f < 48) {
      int l = f >> 4, k = f & 15;
      float gamma = (l == 0) ? pgs[0] : (l == 1) ? pgp[0] : pgd[0];
      float g = fmaxf(gamma, 0.f);
      float ee = expf(-g * r);
      float bern = c_binom[k] * powf(ee, (float)k) * powf(1.f - ee, (float)(15 - k));
      float fcut = 0.f;
      if (r < 5.f) {
        float safe = (5.f - r) * (5.f + r);
        fcut = expf(-r * r / safe);
      }
      rho[half][l][k] = bern * fcut;
    } else {
      int m = f - 48;  // 0..2 -> Yp, 3..7 -> Yd
      float v;
      if (m == 0) v = dy / r;
      else if (m == 1) v = dz / r;
      else if (m == 2) v = dx / r;
      else {
        const float s3 = 1.7320508075688772f;
        float num;
        if (m == 3) num = s3 * dx * dy;
        else if (m == 4) num = s3 * dy * dz;
        else if (m == 5) num = 1.5f * dz * dz - 0.5f * r2;
        else if (m == 6) num = s3 * dx * dz;
        else num = 0.5f * s3 * (dx * dx - dy * dy);
        v = num / r2;
      }
      ysh[half][m] = v;
    }
  }
  __syncthreads();
  if (e < NE) {
    float grs = 0.f, grp = 0.f, grd = 0.f;
#pragma unroll
    for (int k = 0; k < 16; k++) {
      grs += Gs[f * 16 + k] * rho[half][0][k];
      grp += Gp[f * 16 + k] * rho[half][1][k];
      grd += Gd[f * 16 + k] * rho[half][2][k];
    }
    float us = res[1 * NA * F + i1 * F + f] * grs;
    float up = res[2 * NA * F + i1 * F + f] * grp;
    float ud = res[3 * NA * F + i1 * F + f] * grd;
    atomicAdd(&qs[i0 * F + f], us);
#pragma unroll
    for (int m = 0; m < 3; m++)
      atomicAdd(&qp[f * (3 * NA) + i0 * 3 + m], up * ysh[half][m]);
#pragma unroll
    for (int m = 0; m < 5; m++)
      atomicAdd(&qd[f * (5 * NA) + i0 * 5 + m], ud * ysh[half][3 + m]);
  }
}

// ---------------------------------------------------------------------------
// inner kernel: inp[n,i] = res0[n,i] + qs[n,i]
//                        + sum_m (P1 q_p)[n,i,m]*(P2 q_p)[n,i,m]
//                        + sum_m (D1 q_d)[n,i,m]*(D2 q_d)[n,i,m]
// WMMA tile: M = feature i (A rows from P/D), N = atom (B cols from q), K = j.
// ---------------------------------------------------------------------------
__global__ __launch_bounds__(256) void inner_kernel(
    const float* __restrict__ P1, const float* __restrict__ P2,
    const float* __restrict__ D1, const float* __restrict__ D2,
    const float* __restrict__ res0, const float* __restrict__ qs,
    const float* __restrict__ qp, const float* __restrict__ qd,
    float* __restrict__ inp) {
  const int abase = blockIdx.x * 16;
  const int t = threadIdx.x;
  const int wave = t >> 5, lane = t & 31;
  const int l15 = lane & 15, hi = lane >> 4;
  const int fb = wave * 16;
  const int irow = fb + l15;                       // A-fragment row (i)
  const int atom = abase + l15;                    // C column (atom)
  const int acol = (atom < NA) ? atom : (NA - 1);  // clamped gather for tail

  float innerAcc[8];
#pragma unroll
  for (int r = 0; r < 8; r++) innerAcc[r] = 0.f;

  // ---- p channel (3 harmonics)
#pragma unroll
  for (int m = 0; m < 3; m++) {
    v8f c1, c2;
#pragma unroll
    for (int r = 0; r < 8; r++) { c1[r] = 0.f; c2[r] = 0.f; }
    const float* bcol = qp + acol * 3 + m;
#pragma unroll 8
    for (int k0 = 0; k0 < F; k0 += 4) {
      int k = k0 + 2 * hi;
      v2f b;
      b.x = bcol[k * (3 * NA)];
      b.y = bcol[(k + 1) * (3 * NA)];
      v2f a1 = *(const v2f*)&P1[irow * F + k];
      v2f a2 = *(const v2f*)&P2[irow * F + k];
      c1 = wmma4(a1, b, c1);
      c2 = wmma4(a2, b, c2);
    }
#pragma unroll
    for (int r = 0; r < 8; r++) innerAcc[r] += c1[r] * c2[r];
  }

  // ---- d channel (5 harmonics)
#pragma unroll
  for (int m = 0; m < 5; m++) {
    v8f c1, c2;
#pragma unroll
    for (int r = 0; r < 8; r++) { c1[r] = 0.f; c2[r] = 0.f; }
    const float* bcol = qd + acol * 5 + m;
#pragma unroll 8
    for (int k0 = 0; k0 < F; k0 += 4) {
      int k = k0 + 2 * hi;
      v2f b;
      b.x = bcol[k * (5 * NA)];
      b.y = bcol[(k + 1) * (5 * NA)];
      v2f a1 = *(const v2f*)&D1[irow * F + k];
      v2f a2 = *(const v2f*)&D2[irow * F + k];
      c1 = wmma4(a1, b, c1);
      c2 = wmma4(a2, b, c2);
    }
#pragma unroll
    for (int r = 0; r < 8; r++) innerAcc[r] += c1[r] * c2[r];
  }

  if (atom < NA) {
#pragma unroll
    for (int r = 0; r < 8; r++) {
      int i = fb + r + 8 * hi;
      int idx = atom * F + i;
      inp[idx] = res0[idx] + qs[idx] + innerAcc[r];
    }
  }
}

// ---------------------------------------------------------------------------
extern "C" void kernel_launch(void* const* d_in, const int* in_sizes, int n_in,
                              void* d_out, int out_size, void* d_ws,
                              size_t ws_size, hipStream_t stream) {
  const float* xyz = (const float*)d_in[0];
  const float* xt = (const float*)d_in[1];
  const int* nbrs = (const int*)d_in[2];
  const float* rmW1 = (const float*)d_in[3];
  const float* rmB1 = (const float*)d_in[4];
  const float* rmW2 = (const float*)d_in[5];
  const float* rmB2 = (const float*)d_in[6];
  const float* rmW3 = (const float*)d_in[7];
  const float* rmB3 = (const float*)d_in[8];
  const float* rlW1 = (const float*)d_in[9];
  const float* rlB1 = (const float*)d_in[10];
  const float* rlW2 = (const float*)d_in[11];
  const float* rlB2 = (const float*)d_in[12];
  const float* rlW3 = (const float*)d_in[13];
  const float* rlB3 = (const float*)d_in[14];
  const float* Gs = (const float*)d_in[15];
  const float* Gp = (const float*)d_in[16];
  const float* Gd = (const float*)d_in[17];
  const float* P1 = (const float*)d_in[18];
  const float* P2 = (const float*)d_in[19];
  const float* D1 = (const float*)d_in[20];
  const float* D2 = (const float*)d_in[21];
  const float* gs = (const float*)d_in[22];
  const float* gp = (const float*)d_in[23];
  const float* gd = (const float*)d_in[24];
  const float* al = (const float*)d_in[25];
  const float* be = (const float*)d_in[26];

  float* ws = (float*)d_ws;
  float* res = ws;                 // 4*NA*F  (stage-A outputs, 4 channels)
  float* qs = res + 4 * NA * F;    // NA*F
  float* qp = qs + NA * F;         // F * 3*NA (feature-major)
  float* qd = qp + 3 * NA * F;     // F * 5*NA (feature-major)
  float* inp = qd + 5 * NA * F;    // NA*F

  // zero the scatter-add accumulators every call (deterministic)
  hipMemsetAsync(qs, 0, (size_t)(NA * F + 3 * NA * F + 5 * NA * F) * sizeof(float),
                 stream);

  // Stage A: 4 channel residual MLPs
  mlp3_kernel<<<dim3(NT, 4), 256, 0, stream>>>(
      xt, rmW1, rmB1, rmW2, rmB2, rmW3, rmB3, al, be, 0, 1, 2, res, F * F, F,
      NA * F, 1);

  // Edge messages + segment-sum via atomics
  edge_kernel<<<NE / 2, 256, 0, stream>>>(xyz, nbrs, res, Gs, Gp, Gd, gs, gp,
                                          gd, qs, qp, qd);

  // Inner products + combine -> inp
  inner_kernel<<<NT, 256, 0, stream>>>(P1, P2, D1, D2, res, qs, qp, qd, inp);

  // Final residual MLP -> output
  mlp3_kernel<<<dim3(NT, 1), 256, 0, stream>>>(
      inp, rlW1, rlB1, rlW2, rlB2, rlW3, rlB3, al, be, 3, 4, 5, (float*)d_out,
      0, 0, 0, 0);
}